// GrapherDGL_420906795277
// MI455X (gfx1250) — compile-verified
//
#include <hip/hip_runtime.h>
#include <math.h>

typedef __attribute__((ext_vector_type(2))) float v2f;
typedef __attribute__((ext_vector_type(8))) float v8f;

#define CCH 128  // channels

// ---------------------------------------------------------------------------
// Wc = W1 - W2   (W is [2C, C] row-major; W1 = rows 0..C-1, W2 = rows C..2C-1)
// ---------------------------------------------------------------------------
__global__ __launch_bounds__(256) void prep_wc_kernel(const float* __restrict__ W,
                                                      float* __restrict__ Wc) {
  int t = blockIdx.x * blockDim.x + threadIdx.x;
  if (t < CCH * CCH) Wc[t] = W[t] - W[t + CCH * CCH];
}

// ---------------------------------------------------------------------------
// M = -inf
// ---------------------------------------------------------------------------
__global__ __launch_bounds__(256) void init_neg_inf_kernel(float* __restrict__ M, int n) {
  int t = blockIdx.x * blockDim.x + threadIdx.x;
  if (t < n) M[t] = -INFINITY;
}

// ---------------------------------------------------------------------------
// Node GEMMs via V_WMMA_F32_16X16X4_F32 (wave32; one wave per 16x16 out tile).
//   tileN 0..7  : Aout = x @ Wc + b
//   tileN 8..15 : Bout = x @ W2
// A-operand (16x4 f32, 2 VGPRs): lane = 16*kHalf + m  holds K = 2*kHalf, 2*kHalf+1
// B-operand (4x16 f32, 2 VGPRs): lane = 16*kHalf + n  holds K = 2*kHalf, 2*kHalf+1
// C/D (16x16 f32, 8 VGPRs): VGPR r -> row r + 8*kHalf, col = lane & 15
// ---------------------------------------------------------------------------
__global__ __launch_bounds__(256) void node_gemm_kernel(
    const float* __restrict__ x, const float* __restrict__ Wc,
    const float* __restrict__ W2, const float* __restrict__ bias,
    float* __restrict__ Aout, float* __restrict__ Bout, int Nn) {
  int wave  = (blockIdx.x * blockDim.x + threadIdx.x) >> 5;
  int lane  = threadIdx.x & 31;
  int tileM = wave >> 4;   // 16 column tiles per row tile (8 for A, 8 for B)
  int tileN = wave & 15;
  int m0 = tileM * 16;
  if (m0 >= Nn) return;  // uniform per wave -> EXEC stays all-ones for WMMA

  bool isA          = (tileN < 8);
  const float* Wp   = isA ? Wc : W2;
  float*       outp = isA ? Aout : Bout;
  int n0 = (tileN & 7) * 16;

  int m     = lane & 15;       // row within tile (A operand / output col index reuse)
  int kHalf = lane >> 4;       // 0 or 1
  int kb    = kHalf * 2;       // K offset held by this lane

  int rowA = m0 + m;
  if (rowA >= Nn) rowA = Nn - 1;           // clamp loads for ragged tail (N=10000: unused)
  const float* xrow = x + (size_t)rowA * CCH;

  v8f acc = {};
  for (int k0 = 0; k0 < CCH; k0 += 4) {
    v2f a, b;
    a.x = xrow[k0 + kb + 0];
    a.y = xrow[k0 + kb + 1];
    b.x = Wp[(size_t)(k0 + kb + 0) * CCH + n0 + m];
    b.y = Wp[(size_t)(k0 + kb + 1) * CCH + n0 + m];
    // 8 args: (neg_a, A, neg_b, B, c_mod, C, reuse_a, reuse_b)
    acc = __builtin_amdgcn_wmma_f32_16x16x4_f32(false, a, false, b,
                                                (short)0, acc, false, false);
  }

  float bv = isA ? bias[n0 + m] : 0.0f;
#pragma unroll
  for (int r = 0; r < 8; ++r) {
    int row = m0 + r + kHalf * 8;
    if (row < Nn) outp[(size_t)row * CCH + n0 + m] = acc[r] + bv;
  }
}

// ---------------------------------------------------------------------------
// Per-edge segment max: M[dst] = max(M[dst], B[src]) channel-wise.
// Sign-split trick -> native non-returning i32/u32 L2 atomics (no CAS loop).
// Total order over IEEE floats: positives as int-max, negatives as uint-min.
// ---------------------------------------------------------------------------
__device__ __forceinline__ void atomicMaxF(float* addr, float v) {
  if (v >= 0.0f)
    atomicMax((int*)addr, __float_as_int(v));
  else
    atomicMin((unsigned int*)addr, __float_as_uint(v));
}

__global__ __launch_bounds__(256) void edge_max_kernel(
    const int* __restrict__ src, const int* __restrict__ dst,
    const float* __restrict__ Bm, float* __restrict__ Mbuf, int E) {
  int t = blockIdx.x * blockDim.x + threadIdx.x;
  int e = t >> 5;            // one wave per edge
  if (e >= E) return;
  int lane = t & 31;         // each lane handles 4 channels

  int s = src[e];
  int d = dst[e];
  const float4* bs = (const float4*)(Bm + (size_t)s * CCH);
  float4 v = bs[lane];       // 16B coalesced, L2-resident
  float* mp = Mbuf + (size_t)d * CCH + lane * 4;
  atomicMaxF(mp + 0, v.x);
  atomicMaxF(mp + 1, v.y);
  atomicMaxF(mp + 2, v.z);
  atomicMaxF(mp + 3, v.w);
}

// ---------------------------------------------------------------------------
// out = isfinite(A + M) ? relu(A + M) : 0   (M == -inf <=> node has no edges)
// ---------------------------------------------------------------------------
__global__ __launch_bounds__(256) void finalize_kernel(
    const float* __restrict__ Aout, const float* __restrict__ Mbuf,
    float* __restrict__ out, int n) {
  int t = blockIdx.x * blockDim.x + threadIdx.x;
  if (t >= n) return;
  float v = Aout[t] + Mbuf[t];
  out[t] = isfinite(v) ? fmaxf(v, 0.0f) : 0.0f;
}

// ---------------------------------------------------------------------------
extern "C" void kernel_launch(void* const* d_in, const int* in_sizes, int n_in,
                              void* d_out, int out_size, void* d_ws, size_t ws_size,
                              hipStream_t stream) {
  const float* x   = (const float*)d_in[0];  // [N, C]
  const float* W   = (const float*)d_in[1];  // [2C, C]
  const float* b   = (const float*)d_in[2];  // [C]
  const int*   src = (const int*)d_in[3];    // [E]
  const int*   dst = (const int*)d_in[4];    // [E]
  float*       out = (float*)d_out;          // [N, C]

  const int N = in_sizes[0] / CCH;
  const int E = in_sizes[3];
  const int NC = N * CCH;

  // workspace layout
  float* Wc   = (float*)d_ws;                // [C, C]   = W1 - W2
  float* Aout = Wc + CCH * CCH;              // [N, C]   = x@Wc + b
  float* Bout = Aout + (size_t)NC;           // [N, C]   = x@W2
  float* Mbuf = Bout + (size_t)NC;           // [N, C]   running segment max

  prep_wc_kernel<<<(CCH * CCH + 255) / 256, 256, 0, stream>>>(W, Wc);
  init_neg_inf_kernel<<<(NC + 255) / 256, 256, 0, stream>>>(Mbuf, NC);

  int tilesM = (N + 15) / 16;
  int nWaves = tilesM * 16;                  // 16 column tiles (8 A + 8 B) per row tile
  int gemmBlocks = (nWaves * 32 + 255) / 256;
  node_gemm_kernel<<<gemmBlocks, 256, 0, stream>>>(x, Wc, W + CCH * CCH, b,
                                                   Aout, Bout, N);

  long long edgeThreads = (long long)E * 32;
  int edgeBlocks = (int)((edgeThreads + 255) / 256);
  edge_max_kernel<<<edgeBlocks, 256, 0, stream>>>(src, dst, Bout, Mbuf, E);

  finalize_kernel<<<(NC + 255) / 256, 256, 0, stream>>>(Aout, Mbuf, out, NC);
}